// CoHeat_39006892982671
// MI455X (gfx1250) — compile-verified
//
#include <hip/hip_runtime.h>
#include <hip/hip_bf16.h>
#include <math.h>
#include <stdint.h>

// ---------------------------------------------------------------------------
// CoHeat propagation for MI455X (gfx1250).
// Memory-bound SpMM pipeline: feature matrices are L2-resident (38.4MB /
// 17.9MB << 192MB L2), so per-edge gather + f32 L2 atomics is the optimal
// data path. Edge-index slabs are staged into LDS via the gfx1250 async
// load-to-LDS engine (ASYNCcnt). wave32 shuffle butterfly for row L2-norms.
// No WMMA: arithmetic intensity ~0.25 flop/byte, far below any matrix-core
// balance point.
// ---------------------------------------------------------------------------

#define COH_U 50000
#define COH_I 100000
#define COH_B 20000
#define COH_D 64
#define COH_EPB 16   // edges per 256-thread block (16 lanes x float4 per edge)

// ---------------------------------------------------------------------------
// SpMM scatter: y[rows[e]] += scale * vals[e] * x[cols[e]]
// 16 lanes per edge, float4 per lane (global_load_b128 / 4x atomic_add_f32).
// Split-source gather: col < split reads xlo, else xhi (lets layer 0 read the
// original users/items buffers without a concat copy).
// Index staging: for full blocks, wave 0 async-copies the block's 16-edge
// (rows|cols|vals) slab (3 x 64B) into LDS with global_load_async_to_lds_b128,
// drains ASYNCcnt, and the block reads indices from LDS broadcast.
// ---------------------------------------------------------------------------
__global__ __launch_bounds__(256) void coheat_spmm_scatter(
    const int* __restrict__ rows, const int* __restrict__ cols,
    const float* __restrict__ vals,
    const float* __restrict__ xlo, const float* __restrict__ xhi, int split,
    float* __restrict__ y, int nE, float scale) {
  __shared__ __align__(16) int sh[3 * COH_EPB];   // rows[16] | cols[16] | vals[16]

  const int e_base = blockIdx.x * COH_EPB;
  const int eib    = threadIdx.x >> 4;            // edge-in-block 0..15
  const int q      = threadIdx.x & 15;            // float4 quarter 0..15
  const bool full  = (e_base + COH_EPB) <= nE;    // block-uniform

  // Stream the edge arrays ahead (gfx1250 global_prefetch_b8).
  if (threadIdx.x == 0 && (e_base + 4096) < nE) {
    __builtin_prefetch(rows + e_base + 4096, 0, 0);
    __builtin_prefetch(cols + e_base + 4096, 0, 0);
    __builtin_prefetch(vals + e_base + 4096, 0, 0);
  }

  if (full && threadIdx.x < 12) {
    // lane -> (array, 16B chunk): 4 chunks per 64B slab, 3 slabs.
    const int arr   = threadIdx.x >> 2;   // 0=rows 1=cols 2=vals
    const int chunk = threadIdx.x & 3;
    const char* gbase = (arr == 0) ? (const char*)rows
                      : (arr == 1) ? (const char*)cols
                                   : (const char*)vals;
    const char* gaddr = gbase + (size_t)e_base * 4 + (size_t)chunk * 16;
    unsigned lds_off = (unsigned)(uintptr_t)(&sh[arr * COH_EPB + chunk * 4]);
    asm volatile("global_load_async_to_lds_b128 %0, %1, off"
                 :: "v"(lds_off), "v"(gaddr) : "memory");
  }
  if (full) {
    asm volatile("s_wait_asynccnt 0x0" ::: "memory");  // wave 0 drains; others no-op
  }
  __syncthreads();

  const int e = e_base + eib;
  if (e >= nE) return;

  int r, c;
  float v;
  if (full) {
    r = sh[eib];
    c = sh[COH_EPB + eib];
    v = __int_as_float(sh[2 * COH_EPB + eib]);
  } else {                                  // single partial tail block
    r = rows[e];
    c = cols[e];
    v = vals[e];
  }
  v *= scale;

  const float* xrow = (c < split) ? (xlo + (size_t)c * COH_D)
                                  : (xhi + (size_t)(c - split) * COH_D);
  float4 f = ((const float4*)xrow)[q];

  float* yo = y + (size_t)r * COH_D + (size_t)q * 4;
  unsafeAtomicAdd(yo + 0, v * f.x);
  unsafeAtomicAdd(yo + 1, v * f.y);
  unsafeAtomicAdd(yo + 2, v * f.z);
  unsafeAtomicAdd(yo + 3, v * f.w);
}

// ---------------------------------------------------------------------------
// Row L2-normalize + accumulate. One row (64 f32) per wave32: 2 floats/lane,
// 5-step __shfl_xor butterfly for the sum of squares.
//   add_mode == 0:  acc[row] = init[row] + x[row] / max(||x[row]||, 1e-12)
//   add_mode == 1:  acc[row] +=            x[row] / max(||x[row]||, 1e-12)
// ---------------------------------------------------------------------------
__global__ __launch_bounds__(256) void coheat_l2norm_accum(
    const float* __restrict__ x, const float* __restrict__ init,
    float* __restrict__ acc, int nrows, int add_mode) {
  int row = blockIdx.x * (blockDim.x >> 5) + (threadIdx.x >> 5);
  int lane = threadIdx.x & 31;
  if (row >= nrows) return;

  float2 f = ((const float2*)(x + (size_t)row * COH_D))[lane];
  float ss = f.x * f.x + f.y * f.y;
#pragma unroll
  for (int off = 16; off > 0; off >>= 1) ss += __shfl_xor(ss, off, 32);

  float inv = 1.0f / fmaxf(sqrtf(ss), 1e-12f);

  float2* ar = (float2*)(acc + (size_t)row * COH_D);
  float2 base;
  if (add_mode) {
    base = ar[lane];
  } else {
    base = ((const float2*)(init + (size_t)row * COH_D))[lane];
  }
  base.x += f.x * inv;
  base.y += f.y * inv;
  ar[lane] = base;
}

// ---------------------------------------------------------------------------

static inline int spmm_blocks(int nE) { return (nE + COH_EPB - 1) / COH_EPB; }
static inline int norm_blocks(int nrows) { return (nrows + 7) / 8; }

extern "C" void kernel_launch(void* const* d_in, const int* in_sizes, int n_in,
                              void* d_out, int out_size, void* d_ws, size_t ws_size,
                              hipStream_t stream) {
  (void)n_in; (void)out_size; (void)ws_size;

  const float* users   = (const float*)d_in[0];   // [U, D]
  const float* items   = (const float*)d_in[1];   // [I, D]
  const float* bundles = (const float*)d_in[2];   // [B, D]
  const int*   aff_rows  = (const int*)  d_in[3];
  const int*   aff_cols  = (const int*)  d_in[4];
  const float* aff_vals  = (const float*)d_in[5];
  const int*   hist_rows = (const int*)  d_in[6];
  const int*   hist_cols = (const int*)  d_in[7];
  const float* hist_vals = (const float*)d_in[8];
  const int*   agg_rows  = (const int*)  d_in[9];
  const int*   agg_cols  = (const int*)  d_in[10];
  const float* agg_vals  = (const float*)d_in[11];

  const int nE_aff  = in_sizes[3];   // 2*NNZ_UI = 3,000,000
  const int nE_hist = in_sizes[6];   // 2*NNZ_UB = 2,000,000
  const int nE_agg  = in_sizes[9];   // NNZ_BI   = 1,500,000

  const long long N1 = (long long)COH_U + COH_I;  // 150000 (aff graph rows)
  const long long N2 = (long long)COH_U + COH_B;  //  70000 (hist graph rows)

  // Workspace: two ping-pong feature buffers sized for the larger graph,
  // plus the aff_items accumulator (input to the agg SpMM).  ~102.4 MB.
  float* ws0  = (float*)d_ws;                 // N1*D
  float* ws1  = ws0 + N1 * COH_D;             // N1*D
  float* accI = ws1 + N1 * COH_D;             // I*D  (aff_items accumulator)

  float* out   = (float*)d_out;
  float* outAU = out;                                          // aff_users   [U,D]
  float* outHU = out + (size_t)COH_U * COH_D;                  // hist_users  [U,D]
  float* outAB = out + (size_t)2 * COH_U * COH_D;              // aff_bundles [B,D]
  float* outHB = outAB + (size_t)COH_B * COH_D;                // hist_bundles[B,D]

  const float s2 = 0.5f, s3 = 1.0f / 3.0f;

  // ============================ AFF phase (U+I) ============================
  // Layer 0: f1 = spmm(concat(users,items)) / 2   (split-source gather)
  hipMemsetAsync(ws1, 0, (size_t)N1 * COH_D * sizeof(float), stream);
  coheat_spmm_scatter<<<spmm_blocks(nE_aff), 256, 0, stream>>>(
      aff_rows, aff_cols, aff_vals, users, items, COH_U, ws1, nE_aff, s2);
  // acc = feats0 + l2norm(f1)   (users part -> d_out, items part -> accI)
  coheat_l2norm_accum<<<norm_blocks(COH_U), 256, 0, stream>>>(
      ws1, users, outAU, COH_U, 0);
  coheat_l2norm_accum<<<norm_blocks(COH_I), 256, 0, stream>>>(
      ws1 + (size_t)COH_U * COH_D, items, accI, COH_I, 0);

  // Layer 1: f2 = spmm(f1) / 3
  hipMemsetAsync(ws0, 0, (size_t)N1 * COH_D * sizeof(float), stream);
  coheat_spmm_scatter<<<spmm_blocks(nE_aff), 256, 0, stream>>>(
      aff_rows, aff_cols, aff_vals, ws1, ws1, 0x7fffffff, ws0, nE_aff, s3);
  coheat_l2norm_accum<<<norm_blocks(COH_U), 256, 0, stream>>>(
      ws0, nullptr, outAU, COH_U, 1);
  coheat_l2norm_accum<<<norm_blocks(COH_I), 256, 0, stream>>>(
      ws0 + (size_t)COH_U * COH_D, nullptr, accI, COH_I, 1);

  // aff_bundles = row-normalized agg graph applied to aff_items accumulator.
  hipMemsetAsync(outAB, 0, (size_t)COH_B * COH_D * sizeof(float), stream);
  coheat_spmm_scatter<<<spmm_blocks(nE_agg), 256, 0, stream>>>(
      agg_rows, agg_cols, agg_vals, accI, accI, 0x7fffffff, outAB, nE_agg, 1.0f);

  // ============================ HIST phase (U+B) ===========================
  hipMemsetAsync(ws1, 0, (size_t)N2 * COH_D * sizeof(float), stream);
  coheat_spmm_scatter<<<spmm_blocks(nE_hist), 256, 0, stream>>>(
      hist_rows, hist_cols, hist_vals, users, bundles, COH_U, ws1, nE_hist, s2);
  coheat_l2norm_accum<<<norm_blocks(COH_U), 256, 0, stream>>>(
      ws1, users, outHU, COH_U, 0);
  coheat_l2norm_accum<<<norm_blocks(COH_B), 256, 0, stream>>>(
      ws1 + (size_t)COH_U * COH_D, bundles, outHB, COH_B, 0);

  hipMemsetAsync(ws0, 0, (size_t)N2 * COH_D * sizeof(float), stream);
  coheat_spmm_scatter<<<spmm_blocks(nE_hist), 256, 0, stream>>>(
      hist_rows, hist_cols, hist_vals, ws1, ws1, 0x7fffffff, ws0, nE_hist, s3);
  coheat_l2norm_accum<<<norm_blocks(COH_U), 256, 0, stream>>>(
      ws0, nullptr, outHU, COH_U, 1);
  coheat_l2norm_accum<<<norm_blocks(COH_B), 256, 0, stream>>>(
      ws0 + (size_t)COH_U * COH_D, nullptr, outHB, COH_B, 1);
}